// GNN_39273180954945
// MI455X (gfx1250) — compile-verified
//
#include <hip/hip_runtime.h>

#define NODES 100000
#define EDGES 600000

typedef __attribute__((ext_vector_type(2))) float v2f;
typedef __attribute__((ext_vector_type(8))) float v8f;

__global__ void zero_kernel(float* __restrict__ p, int n) {
  int i = blockIdx.x * blockDim.x + threadIdx.x;
  if (i < n) p[i] = 0.0f;
}

__global__ void degree_kernel(const long long* __restrict__ dst, float* __restrict__ deg) {
  int e = blockIdx.x * blockDim.x + threadIdx.x;
  if (e < EDGES) {
    unsafeAtomicAdd(&deg[(int)dst[e]], 1.0f);
  }
}

// One wave (32 lanes) per edge; each lane moves 4 floats of the 128-wide feature row.
__global__ void scatter_kernel(const float* __restrict__ h,
                               const long long* __restrict__ src,
                               const long long* __restrict__ dst,
                               float* __restrict__ agg) {
  int gid = blockIdx.x * blockDim.x + threadIdx.x;
  int e = gid >> 5;
  int lane = gid & 31;
  if (e >= EDGES) return;
  int s = (int)src[e];
  int d = (int)dst[e];
  const float4 v = *(const float4*)(h + (size_t)s * 128 + lane * 4);
  float* o = agg + (size_t)d * 128 + lane * 4;
  unsafeAtomicAdd(o + 0, v.x);
  unsafeAtomicAdd(o + 1, v.y);
  unsafeAtomicAdd(o + 2, v.z);
  unsafeAtomicAdd(o + 3, v.w);
}

// Fused SAGE update for a 16-row node tile, one wave32 per block:
//   out[16 x FOUT] = (agg/deg)[16x128] * Wl^T + h[16x128] * Wr^T + bl  (optionally ReLU)
// Uses V_WMMA_F32_16X16X4_F32 (exact fp32). A-fragment: lane L holds row M=L&15,
// K = {k, k+1} (lanes 0-15) or {k+2, k+3} (lanes 16-31). B-fragment mirrors with
// N = L&15. C/D: VGPR j holds (M = j + 8*(L>>4), N = L&15).
template <int FOUT, bool RELU>
__global__ void sage_gemm_kernel(const float* __restrict__ h,
                                 const float* __restrict__ agg,
                                 const float* __restrict__ deg,
                                 const float* __restrict__ Wl,
                                 const float* __restrict__ bl,
                                 const float* __restrict__ Wr,
                                 float* __restrict__ out) {
  constexpr int FIN = 128;
  constexpr int NCT = FOUT / 16;
  const int r0 = blockIdx.x * 16;
  const int lane = threadIdx.x;   // 0..31
  const int mrow = lane & 15;
  const int khalf = lane >> 4;    // 0 or 1
  const int row = r0 + mrow;

  const float inv = 1.0f / fmaxf(deg[row], 1.0f);

  v8f c[NCT] = {};

  const float* arow = agg + (size_t)row * FIN;
  const float* hrow = h + (size_t)row * FIN;

  for (int k = 0; k < FIN; k += 4) {
    const int ka = k + 2 * khalf;
    v2f am = *(const v2f*)(arow + ka);
    am = am * inv;                               // mean aggregation
    const v2f ah = *(const v2f*)(hrow + ka);
#pragma unroll
    for (int t = 0; t < NCT; ++t) {
      const int col = t * 16 + mrow;
      const v2f blv = *(const v2f*)(Wl + (size_t)col * FIN + ka);
      const v2f brv = *(const v2f*)(Wr + (size_t)col * FIN + ka);
      c[t] = __builtin_amdgcn_wmma_f32_16x16x4_f32(false, am, false, blv,
                                                   (short)0, c[t], false, false);
      c[t] = __builtin_amdgcn_wmma_f32_16x16x4_f32(false, ah, false, brv,
                                                   (short)0, c[t], false, false);
    }
  }

#pragma unroll
  for (int t = 0; t < NCT; ++t) {
    const int col = t * 16 + mrow;
    const float bias = bl[col];
#pragma unroll
    for (int j = 0; j < 8; ++j) {
      float v = c[t][j] + bias;
      if (RELU) v = fmaxf(v, 0.0f);
      const int orow = r0 + j + 8 * khalf;
      out[(size_t)orow * FOUT + col] = v;
    }
  }
}

extern "C" void kernel_launch(void* const* d_in, const int* in_sizes, int n_in,
                              void* d_out, int out_size, void* d_ws, size_t ws_size,
                              hipStream_t stream) {
  (void)in_sizes; (void)n_in; (void)out_size; (void)ws_size;

  const float* x = (const float*)d_in[0];
  const long long* ei = (const long long*)d_in[1];
  const long long* src = ei;          // edge_index[0]
  const long long* dst = ei + EDGES;  // edge_index[1]
  const float* Wl0 = (const float*)d_in[2];
  const float* bl0 = (const float*)d_in[3];
  const float* Wr0 = (const float*)d_in[4];
  const float* Wl1 = (const float*)d_in[5];
  const float* bl1 = (const float*)d_in[6];
  const float* Wr1 = (const float*)d_in[7];
  const float* Wl2 = (const float*)d_in[8];
  const float* bl2 = (const float*)d_in[9];
  const float* Wr2 = (const float*)d_in[10];
  float* out = (float*)d_out;

  // Workspace: two N x 128 feature buffers + degree array (~102.8 MB total).
  float* bufA = (float*)d_ws;
  float* bufB = bufA + (size_t)NODES * 128;
  float* deg  = bufB + (size_t)NODES * 128;

  const int feat_elems = NODES * 128;
  dim3 tb(256);
  dim3 zg_feat((feat_elems + 255) / 256);
  dim3 zg_deg((NODES + 255) / 256);
  dim3 eg((EDGES + 255) / 256);
  dim3 sg((EDGES * 32 + 255) / 256);
  dim3 gg(NODES / 16);   // 6250 tiles, N divisible by 16
  dim3 gb(32);           // one wave32 per tile (EXEC all ones for WMMA)

  // Degrees are identical for all three layers: compute once per call.
  zero_kernel<<<zg_deg, tb, 0, stream>>>(deg, NODES);
  degree_kernel<<<eg, tb, 0, stream>>>(dst, deg);

  // Layer 0: agg(x) -> bufA; out -> bufB (ReLU)
  zero_kernel<<<zg_feat, tb, 0, stream>>>(bufA, feat_elems);
  scatter_kernel<<<sg, tb, 0, stream>>>(x, src, dst, bufA);
  sage_gemm_kernel<128, true><<<gg, gb, 0, stream>>>(x, bufA, deg, Wl0, bl0, Wr0, bufB);

  // Layer 1: agg(bufB) -> bufA; out -> bufA in-place (each block only touches its own rows)
  zero_kernel<<<zg_feat, tb, 0, stream>>>(bufA, feat_elems);
  scatter_kernel<<<sg, tb, 0, stream>>>(bufB, src, dst, bufA);
  sage_gemm_kernel<128, true><<<gg, gb, 0, stream>>>(bufB, bufA, deg, Wl1, bl1, Wr1, bufA);

  // Layer 2: agg(bufA) -> bufB; out -> d_out (no ReLU, FOUT=64)
  zero_kernel<<<zg_feat, tb, 0, stream>>>(bufB, feat_elems);
  scatter_kernel<<<sg, tb, 0, stream>>>(bufA, src, dst, bufB);
  sage_gemm_kernel<64, false><<<gg, gb, 0, stream>>>(bufA, bufB, deg, Wl2, bl2, Wr2, out);
}